// ECPEC_35691178230110
// MI455X (gfx1250) — compile-verified
//
#include <hip/hip_runtime.h>
#include <math.h>

typedef __attribute__((ext_vector_type(2))) float v2f;
typedef __attribute__((ext_vector_type(8))) float v8f;

namespace {
constexpr int Bn   = 2048;
constexpr int Ln   = 128;
constexpr int Dn   = 100;   // feature dim (=2H)
constexpr int Hn   = 50;
constexpr int G4   = 4 * Hn;   // 200 gates
constexpr int NEn  = 1024;
constexpr int NCn  = 1024;
constexpr int NPAD = 112;      // padded N: 100 j's + gram col (j==100) + zero pad
}

__device__ __forceinline__ float sigf(float x) { return 1.0f / (1.0f + expf(-x)); }

// ---------------------------------------------------------------------------
// Kernel 1: per-batch-row LSTM recurrence, fully resident in LDS.
// Block b owns row b: loops t = 0..len-1 (mask t<len == only update while valid),
// then runs the single backward cell on x[b, len-1] (h0=c0=0 -> W_hh_b unused).
// Weights stored transposed [d][j] in LDS for conflict-free broadcast reads.
// ---------------------------------------------------------------------------
__global__ void ecpec_lstm_kernel(
    const float* __restrict__ we, const int* __restrict__ lens,
    const float* __restrict__ Wihf, const float* __restrict__ Whhf,
    const float* __restrict__ bihf, const float* __restrict__ bhhf,
    const float* __restrict__ Wihb, const float* __restrict__ bihb,
    const float* __restrict__ bhhb, float* __restrict__ U)
{
  extern __shared__ float s[];
  float* WihT  = s;           // 100 x 200 (d-major)
  float* WhhT  = s + 20000;   // 50 x 200
  float* biasF = s + 30000;   // 200
  float* biasB = s + 30200;   // 200
  float* xbuf  = s + 30400;   // 100
  float* gbuf  = s + 30500;   // 200
  float* hbuf  = s + 30700;   // 50
  float* cbuf  = s + 30750;   // 50

  const int tid = threadIdx.x;
  const int b   = blockIdx.x;

  for (int i = tid; i < G4 * Dn; i += 256) {
    int j = i / Dn, d = i - j * Dn;
    WihT[d * G4 + j] = Wihf[i];
  }
  for (int i = tid; i < G4 * Hn; i += 256) {
    int j = i / Hn, d = i - j * Hn;
    WhhT[d * G4 + j] = Whhf[i];
  }
  if (tid < G4) {
    biasF[tid] = bihf[tid] + bhhf[tid];
    biasB[tid] = bihb[tid] + bhhb[tid];
  }
  if (tid < Hn) { hbuf[tid] = 0.0f; cbuf[tid] = 0.0f; }
  __syncthreads();

  const int len = lens[b];
  const float* xrow = we + ((size_t)b * Ln) * Dn;

  for (int t = 0; t < len; ++t) {
    if (tid < Dn) xbuf[tid] = xrow[t * Dn + tid];
    __syncthreads();
    if (tid < G4) {
      float g = biasF[tid];
      #pragma unroll 4
      for (int d = 0; d < Dn; ++d) g += xbuf[d] * WihT[d * G4 + tid];
      #pragma unroll 2
      for (int d = 0; d < Hn; ++d) g += hbuf[d] * WhhT[d * G4 + tid];
      gbuf[tid] = g;
    }
    __syncthreads();
    if (tid < Hn) {
      float gi = gbuf[tid], gf = gbuf[Hn + tid];
      float gg = gbuf[2 * Hn + tid], go = gbuf[3 * Hn + tid];
      float c = sigf(gf) * cbuf[tid] + sigf(gi) * tanhf(gg);
      float h = sigf(go) * tanhf(c);
      cbuf[tid] = c; hbuf[tid] = h;
    }
    __syncthreads();
  }

  // Backward single cell on x_last (still in xbuf from t=len-1); h0=c0=0.
  for (int i = tid; i < G4 * Dn; i += 256) {
    int j = i / Dn, d = i - j * Dn;
    WihT[d * G4 + j] = Wihb[i];
  }
  __syncthreads();
  if (tid < G4) {
    float g = biasB[tid];
    #pragma unroll 4
    for (int d = 0; d < Dn; ++d) g += xbuf[d] * WihT[d * G4 + tid];
    gbuf[tid] = g;
  }
  __syncthreads();
  if (tid < Hn) {
    float gi = gbuf[tid], gg = gbuf[2 * Hn + tid], go = gbuf[3 * Hn + tid];
    float c  = sigf(gi) * tanhf(gg);       // sig(f)*c0 == 0
    float hb = sigf(go) * tanhf(c);
    U[(size_t)b * Dn + tid]      = hbuf[tid];
    U[(size_t)b * Dn + Hn + tid] = hb;
  }
}

// ---------------------------------------------------------------------------
// Kernel 2: per-row projections PE = E@W1^T, PC = C@W2^T and squared norms.
// Row r of U: even -> E row r/2, odd -> C row r/2.
// ---------------------------------------------------------------------------
__global__ void ecpec_proj_kernel(const float* __restrict__ U, const float* __restrict__ W3w,
                                  float* __restrict__ PE, float* __restrict__ PC,
                                  float* __restrict__ nE, float* __restrict__ nC)
{
  __shared__ float urow[Dn];
  const int r   = blockIdx.x;
  const int tid = threadIdx.x;
  if (tid < Dn) urow[tid] = U[(size_t)r * Dn + tid];
  __syncthreads();
  const int  half = r >> 1;
  const bool even = (r & 1) == 0;
  const int  off  = even ? 0 : Dn;
  if (tid < Dn) {
    float acc = 0.0f;
    #pragma unroll 4
    for (int d = 0; d < Dn; ++d) acc += urow[d] * W3w[tid * 301 + off + d];
    (even ? PE : PC)[half * Dn + tid] = acc;
  }
  if (tid == 0) {
    float n = 0.0f;
    for (int d = 0; d < Dn; ++d) n += urow[d] * urow[d];
    (even ? nE : nC)[half] = n;
  }
}

// ---------------------------------------------------------------------------
// Kernel 3: pairwise MLP via fp32 WMMA (V_WMMA_F32_16X16X4_F32).
// Block = 8 waves = 16 e's x 8 c's. Each wave: M=16 pairs (e varies, c fixed),
// A[p,d] = E[e0+p,d]*C[c,d] built on the fly, B = [W4^T | ones | 0] (100x112)
// stored K-paired so each B fragment is one aligned ds_load_b64.
// Gram (E.C) falls out of the ones column (j==100) -> item2 for free.
// __launch_bounds__(256, 1): max VGPR budget, no accumulator spills.
// ---------------------------------------------------------------------------
__global__ void __launch_bounds__(256, 1) ecpec_pair_kernel(
                                  const float* __restrict__ U,
                                  const float* __restrict__ PE, const float* __restrict__ PC,
                                  const float* __restrict__ nE, const float* __restrict__ nC,
                                  const float* __restrict__ W3w, const float* __restrict__ W3b,
                                  const float* __restrict__ clsw, const float* __restrict__ clsb,
                                  float* __restrict__ out)
{
  extern __shared__ float s[];
  float* Wb  = s;          // 50 x 112 x 2: [(d/2)][j] -> (B[d][j], B[d+1][j]) pairs
  float* Et  = s + 11200;  // 16 x 100
  float* Ct  = s + 12800;  // 8 x 100
  float* PEt = s + 13600;  // 16 x 112 (zero padded)
  float* PCt = s + 15392;  // 8 x 112
  float* WnB = s + 16288;  // 112 (item2 weight column)
  float* bB  = s + 16400;  // 112
  float* c0B = s + 16512;  // 112
  float* c1B = s + 16624;  // 112
  float* nEt = s + 16736;  // 16
  float* nCt = s + 16752;  // 8
  float* gT  = s + 16760;  // 8 waves x 16 pairs (gram values)

  const int tid = threadIdx.x;
  const int e0  = blockIdx.x * 16;
  const int c0  = blockIdx.y * 8;

  // B = [W4^T | ones | 0], K-paired: flat i = ((d>>1)*NPAD + j)*2 + (d&1)
  for (int i = tid; i < (Dn / 2) * NPAD * 2; i += 256) {
    int parity = i & 1;
    int idx2   = i >> 1;
    int q      = idx2 / NPAD;
    int j      = idx2 - q * NPAD;
    int d      = 2 * q + parity;
    Wb[i] = (j < Dn) ? W3w[j * 301 + 201 + d] : (j == Dn ? 1.0f : 0.0f);
  }
  for (int i = tid; i < 16 * Dn; i += 256) {
    int p = i / Dn, d = i - p * Dn;
    Et[i] = U[(size_t)(2 * (e0 + p)) * Dn + d];
  }
  for (int i = tid; i < 8 * Dn; i += 256) {
    int c = i / Dn, d = i - c * Dn;
    Ct[i] = U[(size_t)(2 * (c0 + c) + 1) * Dn + d];
  }
  for (int i = tid; i < 16 * NPAD; i += 256) {
    int p = i / NPAD, j = i - p * NPAD;
    PEt[i] = (j < Dn) ? PE[(e0 + p) * Dn + j] : 0.0f;
  }
  for (int i = tid; i < 8 * NPAD; i += 256) {
    int c = i / NPAD, j = i - c * NPAD;
    PCt[i] = (j < Dn) ? PC[(c0 + c) * Dn + j] : 0.0f;
  }
  if (tid < NPAD) {
    WnB[tid] = (tid < Dn) ? W3w[tid * 301 + 200] : 0.0f;
    bB[tid]  = (tid < Dn) ? W3b[tid] : 0.0f;
    c0B[tid] = (tid < Dn) ? clsw[tid] : 0.0f;
    c1B[tid] = (tid < Dn) ? clsw[Dn + tid] : 0.0f;
  }
  if (tid < 16) nEt[tid] = nE[e0 + tid];
  if (tid < 8)  nCt[tid] = nC[c0 + tid];
  __syncthreads();

  const int w    = tid >> 5;   // wave id == local cause index
  const int lane = tid & 31;
  const int hi   = lane >> 4;  // half-wave
  const int lm   = lane & 15;

  v8f zero = {};
  v8f acc[7];
  #pragma unroll
  for (int nt = 0; nt < 7; ++nt) acc[nt] = zero;

  const float* EtRow = Et + lm * Dn;   // A-matrix M index = lm (ISA 16x4 f32 layout)
  const float* CtRow = Ct + w * Dn;

  #pragma unroll
  for (int k4 = 0; k4 < Dn; k4 += 4) {
    const int dA = k4 + (hi << 1);     // lanes 0-15: K=k4,k4+1 ; lanes 16-31: K=k4+2,k4+3
    v2f ev = *(const v2f*)(EtRow + dA);     // 8B-aligned (dA even, row stride 400B)
    v2f cv = *(const v2f*)(CtRow + dA);
    v2f a  = ev * cv;                       // v_pk_mul_f32
    const v2f* brow = (const v2f*)Wb + ((dA >> 1) * NPAD + lm);
    #pragma unroll
    for (int nt = 0; nt < 7; ++nt) {
      v2f bv = brow[nt * 16];               // one ds_load_b64, even-aligned pair
      acc[nt] = __builtin_amdgcn_wmma_f32_16x16x4_f32(
          false, a, false, bv, (short)0, acc[nt], false, false);
    }
  }

  // Gram column j==100 sits in tile nt=6, column lm==4 (C/D layout: M = v + 8*hi).
  if (lm == 4) {
    #pragma unroll
    for (int v = 0; v < 8; ++v) gT[w * 16 + v + (hi << 3)] = acc[6][v];
  }
  __syncthreads();

  float it2v[8];
  #pragma unroll
  for (int v = 0; v < 8; ++v) {
    int p = v + (hi << 3);
    float g = gT[w * 16 + p];
    it2v[v] = sqrtf(fmaxf(nEt[p] + nCt[w] - 2.0f * g, 0.0f));
  }

  float s0[8], s1[8];
  #pragma unroll
  for (int v = 0; v < 8; ++v) { s0[v] = 0.0f; s1[v] = 0.0f; }

  #pragma unroll
  for (int nt = 0; nt < 7; ++nt) {
    int j = nt * 16 + lm;
    float wn = WnB[j], bb = bB[j], cw0 = c0B[j], cw1 = c1B[j];
    #pragma unroll
    for (int v = 0; v < 8; ++v) {
      int p = v + (hi << 3);
      float pre  = acc[nt][v] + PEt[p * NPAD + j] + PCt[w * NPAD + j] + bb + wn * it2v[v];
      float hval = 1.0f / (1.0f + expf(-pre));
      s0[v] += hval * cw0;   // cls weights are zero-padded for j >= 100
      s1[v] += hval * cw1;
    }
  }

  // Reduce logits over the 16 j-lanes of each half-wave (wave32, width-16 xor).
  #pragma unroll
  for (int off = 8; off >= 1; off >>= 1) {
    #pragma unroll
    for (int v = 0; v < 8; ++v) {
      s0[v] += __shfl_xor(s0[v], off, 16);
      s1[v] += __shfl_xor(s1[v], off, 16);
    }
  }

  if (lm == 0) {
    const float cb0 = clsb[0], cb1 = clsb[1];
    #pragma unroll
    for (int v = 0; v < 8; ++v) {
      int p = v + (hi << 3);
      size_t pairIdx = (size_t)(e0 + p) * NCn + (c0 + w);
      float lg0 = s0[v] + cb0, lg1 = s1[v] + cb1;
      float m  = fmaxf(lg0, lg1);
      float lz = logf(expf(lg0 - m) + expf(lg1 - m));
      out[2 * pairIdx]     = lg0 - m - lz;
      out[2 * pairIdx + 1] = lg1 - m - lz;
    }
  }
}

// ---------------------------------------------------------------------------
// Kernel 4: label_pair[e*1024+c] = any_k(label3[e,k] == c)
// ---------------------------------------------------------------------------
__global__ void ecpec_label_kernel(const int* __restrict__ lab, float* __restrict__ out2)
{
  int idx = blockIdx.x * 256 + threadIdx.x;
  if (idx >= NEn * NCn) return;
  int e = idx >> 10;
  int c = idx & 1023;
  const int* l = lab + e * 3;
  out2[idx] = (l[0] == c || l[1] == c || l[2] == c) ? 1.0f : 0.0f;
}

extern "C" void kernel_launch(void* const* d_in, const int* in_sizes, int n_in,
                              void* d_out, int out_size, void* d_ws, size_t ws_size,
                              hipStream_t stream)
{
  (void)in_sizes; (void)n_in; (void)out_size; (void)ws_size;

  const float* we   = (const float*)d_in[0];   // word_encode (B,L,D)
  const int*   lens = (const int*)d_in[1];     // sen_lengths (B,1)
  const int*   lab  = (const int*)d_in[4];     // label3 (B/2,1,3)
  const float* Wihf = (const float*)d_in[5];
  const float* Whhf = (const float*)d_in[6];
  const float* bihf = (const float*)d_in[7];
  const float* bhhf = (const float*)d_in[8];
  const float* Wihb = (const float*)d_in[9];
  const float* bihb = (const float*)d_in[11];  // W_hh_b (d_in[10]) unused: h0 == 0
  const float* bhhb = (const float*)d_in[12];
  const float* W3w  = (const float*)d_in[13];
  const float* W3b  = (const float*)d_in[14];
  const float* clsw = (const float*)d_in[15];
  const float* clsb = (const float*)d_in[16];
  float* out = (float*)d_out;

  float* ws = (float*)d_ws;                 // ~1.65 MB of fp32 scratch
  float* U  = ws;                           // 2048 x 100
  float* PE = ws + 204800;                  // 1024 x 100
  float* PC = ws + 307200;                  // 1024 x 100
  float* nE = ws + 409600;                  // 1024
  float* nC = ws + 410624;                  // 1024

  ecpec_lstm_kernel<<<Bn, 256, 30800 * sizeof(float), stream>>>(
      we, lens, Wihf, Whhf, bihf, bhhf, Wihb, bihb, bhhb, U);
  ecpec_proj_kernel<<<Bn, 128, 0, stream>>>(U, W3w, PE, PC, nE, nC);
  ecpec_pair_kernel<<<dim3(64, 128), 256, 16888 * sizeof(float), stream>>>(
      U, PE, PC, nE, nC, W3w, W3b, clsw, clsb, out);
  ecpec_label_kernel<<<4096, 256, 0, stream>>>(lab, out + 2 * 1024 * 1024);
}